// OrthMapLayer_23089744183356
// MI455X (gfx1250) — compile-verified
//
#include <hip/hip_runtime.h>

typedef __attribute__((ext_vector_type(2))) float v2f;
typedef __attribute__((ext_vector_type(8))) float v8f;

#define N        64
#define NPAD     65
#define PCOLS    16
#define NSWEEPS  10
#define THREADS  256

// One workgroup per 64x64 symmetric matrix.
// Phase 1: parallel cyclic Jacobi in LDS (A diagonalized, V accumulates eigenvectors).
// Phase 2: rank eigenvalues ascending -> permutation.
// Phase 3: Out = V * P (P = column-selection of eigvec columns 1..16) on the
//          matrix engine via V_WMMA_F32_16X16X4_F32 (4 row-tiles, 1 per wave).
__global__ __launch_bounds__(THREADS)
void orthmap_jacobi_eigh_kernel(const float* __restrict__ x, float* __restrict__ out)
{
    __shared__ float As[N][NPAD];    // working symmetric matrix
    __shared__ float Vs[N][NPAD];    // accumulated eigenvectors (columns)
    __shared__ float Ps[N][PCOLS + 1]; // selection matrix, padded
    __shared__ float cS[32], sS[32];
    __shared__ int   pS[32], qS[32];
    __shared__ int   perm[N];

    const int tid = threadIdx.x;
    const int m   = blockIdx.x;
    const float* __restrict__ src = x + (size_t)m * (N * N);

    // ---- load A, init V = I (coalesced global, conflict-free LDS) ----
    for (int t = tid; t < N * N; t += THREADS) {
        const int r0 = t >> 6, c0 = t & 63;
        As[r0][c0] = src[t];
        Vs[r0][c0] = (r0 == c0) ? 1.0f : 0.0f;
    }
    __syncthreads();

    // ---- parallel cyclic Jacobi sweeps ----
    for (int sweep = 0; sweep < NSWEEPS; ++sweep) {
        for (int r = 0; r < N - 1; ++r) {
            // 32 disjoint pairs per round (circle / tournament schedule)
            if (tid < 32) {
                int p, q;
                if (tid == 0) { p = N - 1; q = r; }
                else {
                    p = (r + tid) % (N - 1);
                    q = (r - tid + (N - 1)) % (N - 1);
                }
                const int lo = p < q ? p : q;
                const int hi = p < q ? q : p;
                const float app = As[lo][lo];
                const float aqq = As[hi][hi];
                const float apq = As[lo][hi];
                float c = 1.0f, s = 0.0f;
                if (fabsf(apq) > 1e-12f) {
                    const float tau = (aqq - app) / (2.0f * apq);
                    const float tt  = copysignf(1.0f, tau) /
                                      (fabsf(tau) + sqrtf(1.0f + tau * tau));
                    c = 1.0f / sqrtf(1.0f + tt * tt);
                    s = tt * c;
                }
                cS[tid] = c; sS[tid] = s; pS[tid] = lo; qS[tid] = hi;
            }
            __syncthreads();

            // rows: A <- J^T A   (32 pairs x 64 cols = 2048 tasks, 8 per thread)
            #pragma unroll
            for (int it = 0; it < (32 * N) / THREADS; ++it) {
                const int t  = tid + it * THREADS;
                const int pr = t >> 6, j = t & 63;
                const int p = pS[pr], q = qS[pr];
                const float c = cS[pr], s = sS[pr];
                const float xr = As[p][j], yr = As[q][j];
                As[p][j] = c * xr - s * yr;
                As[q][j] = s * xr + c * yr;
            }
            __syncthreads();

            // cols: A <- A J  and  V <- V J   (stride-65 rows => conflict-free)
            #pragma unroll
            for (int it = 0; it < (32 * N) / THREADS; ++it) {
                const int t  = tid + it * THREADS;
                const int pr = t >> 6, j = t & 63;
                const int p = pS[pr], q = qS[pr];
                const float c = cS[pr], s = sS[pr];
                float xc = As[j][p], yc = As[j][q];
                As[j][p] = c * xc - s * yc;
                As[j][q] = s * xc + c * yc;
                xc = Vs[j][p]; yc = Vs[j][q];
                Vs[j][p] = c * xc - s * yc;
                Vs[j][q] = s * xc + c * yc;
            }
            __syncthreads();
        }
    }

    // ---- rank eigenvalues ascending (stable tiebreak) ----
    if (tid < N) {
        const float d = As[tid][tid];
        int rank = 0;
        for (int j = 0; j < N; ++j) {
            const float dj = As[j][j];
            rank += (dj < d || (dj == d && j < tid)) ? 1 : 0;
        }
        perm[rank] = tid;
    }
    __syncthreads();

    // ---- build selection matrix P: P[j][k] = (j == perm[k+1]) ----
    for (int t = tid; t < N * PCOLS; t += THREADS)
        Ps[t >> 4][t & 15] = 0.0f;
    __syncthreads();
    if (tid < PCOLS)
        Ps[perm[tid + 1]][tid] = 1.0f;
    __syncthreads();

    // ---- Out = V * P via V_WMMA_F32_16X16X4_F32, one 16-row tile per wave ----
    const int wave = tid >> 5;
    const int lane = tid & 31;
    if (wave < 4) {
        const int  rowbase = wave * 16;
        const int  l       = lane & 15;
        const bool hiHalf  = lane >= 16;
        v8f acc = {};
        for (int k0 = 0; k0 < N; k0 += 4) {
            // A tile (16x4 of V): v0 = K{0|2}, v1 = K{1|3} per ISA layout
            const int ka0 = k0 + (hiHalf ? 2 : 0);
            const int ka1 = k0 + (hiHalf ? 3 : 1);
            v2f a, b;
            a[0] = Vs[rowbase + l][ka0];
            a[1] = Vs[rowbase + l][ka1];
            // B tile (4x16 of P): rows striped across lanes, same K split
            b[0] = Ps[ka0][l];
            b[1] = Ps[ka1][l];
            acc = __builtin_amdgcn_wmma_f32_16x16x4_f32(
                /*neg_a=*/false, a, /*neg_b=*/false, b,
                /*c_mod=*/(short)0, acc, /*reuse_a=*/false, /*reuse_b=*/false);
        }
        // C/D layout: VGPR r -> M = r (lanes 0-15) / M = 8+r (lanes 16-31), N = lane%16
        float* __restrict__ dst = out + (size_t)m * (N * PCOLS);
        #pragma unroll
        for (int rr = 0; rr < 8; ++rr) {
            const int row = rowbase + rr + (hiHalf ? 8 : 0);
            dst[row * PCOLS + l] = acc[rr];
        }
    }
}

extern "C" void kernel_launch(void* const* d_in, const int* in_sizes, int n_in,
                              void* d_out, int out_size, void* d_ws, size_t ws_size,
                              hipStream_t stream)
{
    (void)n_in; (void)d_ws; (void)ws_size; (void)out_size;
    const float* x  = (const float*)d_in[0];
    float*       out = (float*)d_out;
    const int nmat = in_sizes[0] / (N * N);   // 512*16 = 8192 matrices
    orthmap_jacobi_eigh_kernel<<<dim3(nmat), dim3(THREADS), 0, stream>>>(x, out);
}